// GramSchmidt_30442728194303
// MI455X (gfx1250) — compile-verified
//
#include <hip/hip_runtime.h>

// Gram-Schmidt over x[B=8,C=32,N=128,D=1024] f32.
// One workgroup per (b,c): block classical Gram-Schmidt with 16-row blocks.
//   Stage:   raw 16x1024 block -> LDS (single read of x_in).
//   Phase A: T[q][p] = dot(Xprior[p], Xraw[q]) * rcp(||Xprior[p]||^2)  (WMMA f32 16x16x4, K=1024,
//            A from global prior rows, B from LDS; cross-wave reduce via LDS)
//   Phase B: Xblk[q][d] -= sum_p T[q][p] * Xprior[p][d]                (WMMA, K=16*prior, in-place LDS)
//   Phase C: sequential CGS inside the 16-row block in LDS; row norms^2 + reciprocals recorded.
//   Final:   per-column (axis=2) normalization with v_rsq.

#define B_  8
#define C_  32
#define N_  128
#define D_  1024
#define BLK 16
#define NBLK (N_ / BLK)
#define NTHREADS 256
#define NWAVES 8
#define KSLICE (D_ / NWAVES)   // 128 K per wave in phase A

typedef __attribute__((ext_vector_type(2))) float v2f;
typedef __attribute__((ext_vector_type(8))) float v8f;

__global__ __launch_bounds__(NTHREADS)
void GramSchmidt_30442728194303_kernel(const float* __restrict__ x_in,
                                       float* __restrict__ x_out) {
    extern __shared__ float lds[];
    float* Xblk    = lds;                    // 16*1024      = 16384 f (64 KB)
    float* Tt      = Xblk + BLK * D_;        // 16*128       =  2048 f ( 8 KB)  Tt[q][p]
    float* Sred    = Tt + BLK * N_;          // 8 waves *256 =  2048 f ( 8 KB)
    float* norms2  = Sred + NWAVES * 256;    // 128 f
    float* rnorms2 = norms2 + N_;            // 128 f
    float* coeff   = rnorms2 + N_;           // 16 f
    float* wsum    = coeff + BLK;            // 8 f

    const int wg   = blockIdx.x;             // 0..255 -> flattened (b,c)
    const float* xin = x_in  + (size_t)wg * N_ * D_;
    float*       X   = x_out + (size_t)wg * N_ * D_;

    const int tid  = threadIdx.x;
    const int lane = tid & 31;
    const int wave = tid >> 5;
    const int lmod = lane & 15;
    const int lhi  = lane >> 4;              // 0 = lanes 0-15, 1 = lanes 16-31
    const int kofs = lhi * 2;                // A/B operand K offset per ISA layout

    for (int blk = 0; blk < NBLK; ++blk) {
        const int r0 = blk * BLK;

        // ---------- Stage: raw block rows x_in -> LDS ----------
        for (int e = tid; e < BLK * D_ / 4; e += NTHREADS)
            ((float4*)Xblk)[e] = ((const float4*)(xin + (size_t)r0 * D_))[e];
        __syncthreads();

        // ---------- Phase A: Tt[q][p] = dot(prior p, raw q) * rnorm2[p] ----------
        for (int pb = 0; pb < blk; ++pb) {
            // lane base pointers: uniform loop counter keeps EXEC all-ones for WMMA
            const float* ApL = X + (size_t)(pb * BLK + lmod) * D_ + wave * KSLICE + kofs;
            const float* BbL = Xblk + lmod * D_ + wave * KSLICE + kofs;
            v8f acc = {0.f, 0.f, 0.f, 0.f, 0.f, 0.f, 0.f, 0.f};
            #pragma unroll 8
            for (int kk = 0; kk < KSLICE; kk += 4) {
                v2f a = *(const v2f*)(ApL + kk);   // global (L2-resident prior rows)
                v2f b = *(const v2f*)(BbL + kk);   // LDS raw block
                acc = __builtin_amdgcn_wmma_f32_16x16x4_f32(
                        false, a, false, b, (short)0, acc, false, false);
            }
            // cross-wave reduction of partial 16x16 S through LDS
            for (int r = 0; r < 8; ++r)
                Sred[wave * 256 + r * 32 + lane] = acc[r];
            __syncthreads();
            {
                float s = 0.f;
                for (int w = 0; w < NWAVES; ++w) s += Sred[w * 256 + tid];
                const int r  = tid >> 5;
                const int ls = tid & 31;
                const int m  = r + (ls >> 4) * 8;   // C/D layout: VGPR r -> M=r / M=r+8
                const int n  = ls & 15;
                const int p  = pb * BLK + m;        // absolute prior-row index
                Tt[n * N_ + p] = s * rnorms2[p];    // transposed for phase-B b64 loads
            }
            __syncthreads();
        }

        // ---------- Phase B: Xblk[q][d] -= sum_p Tt[q][p] * Xprior[p][d] ----------
        if (r0 > 0) {
            for (int t = wave; t < D_ / 16; t += NWAVES) {
                const int d0 = t * 16;
                const float* TtL = Tt + lmod * N_ + kofs;
                v8f acc = {0.f, 0.f, 0.f, 0.f, 0.f, 0.f, 0.f, 0.f};
                #pragma unroll 2
                for (int k = 0; k < r0; k += 4) {
                    v2f a = *(const v2f*)(TtL + k);  // ds_load_b64: A[q][kk] = Tt[q][k+kk]
                    v2f b;                           // B[kk][n] = Xprior[k+kk][d0+n]
                    b.x = X[(size_t)(k + kofs    ) * D_ + d0 + lmod];
                    b.y = X[(size_t)(k + kofs + 1) * D_ + d0 + lmod];
                    acc = __builtin_amdgcn_wmma_f32_16x16x4_f32(
                            false, a, false, b, (short)0, acc, false, false);
                }
                for (int r = 0; r < 8; ++r) {
                    const int q  = r + lhi * 8;
                    const int dd = d0 + lmod;
                    Xblk[q * D_ + dd] -= acc[r];
                }
            }
            __syncthreads();
        }

        // ---------- Phase C: sequential CGS inside the block (in LDS) ----------
        for (int i = 0; i < BLK; ++i) {
            if (i > 0) {
                // dots(row i, row j) for j < i: 16 threads per j, 64 cols each
                const int j  = tid >> 4;
                const int cs = (tid & 15) * (D_ / 16);
                float s = 0.f;
                if (j < i)
                    for (int d = 0; d < D_ / 16; ++d)
                        s += Xblk[i * D_ + cs + d] * Xblk[j * D_ + cs + d];
                for (int off = 8; off > 0; off >>= 1)
                    s += __shfl_down(s, off, 16);
                if ((tid & 15) == 0 && j < i)
                    coeff[j] = s * rnorms2[r0 + j];
                __syncthreads();
                // subtract correction from row i (4 cols per thread)
                {
                    const int d = tid * 4;
                    float c0 = 0.f, c1 = 0.f, c2 = 0.f, c3 = 0.f;
                    for (int j2 = 0; j2 < i; ++j2) {
                        const float cf = coeff[j2];
                        c0 += cf * Xblk[j2 * D_ + d];
                        c1 += cf * Xblk[j2 * D_ + d + 1];
                        c2 += cf * Xblk[j2 * D_ + d + 2];
                        c3 += cf * Xblk[j2 * D_ + d + 3];
                    }
                    Xblk[i * D_ + d]     -= c0;
                    Xblk[i * D_ + d + 1] -= c1;
                    Xblk[i * D_ + d + 2] -= c2;
                    Xblk[i * D_ + d + 3] -= c3;
                }
                __syncthreads();
            }
            // ||row i||^2 and its reciprocal
            {
                const int d = tid * 4;
                float s = 0.f;
                for (int kk = 0; kk < 4; ++kk) {
                    const float v = Xblk[i * D_ + d + kk];
                    s += v * v;
                }
                for (int off = 16; off > 0; off >>= 1)
                    s += __shfl_down(s, off, 32);
                if (lane == 0) wsum[wave] = s;
                __syncthreads();
                if (tid == 0) {
                    float t2 = 0.f;
                    for (int w = 0; w < NWAVES; ++w) t2 += wsum[w];
                    norms2[r0 + i]  = t2;
                    rnorms2[r0 + i] = __builtin_amdgcn_rcpf(t2);  // v_rcp_f32
                }
                __syncthreads();
            }
        }

        // ---------- write finalized block to x_out ----------
        for (int e = tid; e < BLK * D_ / 4; e += NTHREADS)
            ((float4*)(X + (size_t)r0 * D_))[e] = ((const float4*)Xblk)[e];
        __syncthreads();
        __threadfence();
    }

    // ---------- Final: normalize over the N axis (per-column norms) ----------
    {
        const int d = tid * 4;   // 256 threads * 4 = 1024 = D
        float s0 = 0.f, s1 = 0.f, s2 = 0.f, s3 = 0.f;
        for (int i = 0; i < N_; ++i) {
            const float4 v = *(const float4*)(X + (size_t)i * D_ + d);
            s0 += v.x * v.x; s1 += v.y * v.y; s2 += v.z * v.z; s3 += v.w * v.w;
        }
        const float r0n = __builtin_amdgcn_rsqf(s0);
        const float r1n = __builtin_amdgcn_rsqf(s1);
        const float r2n = __builtin_amdgcn_rsqf(s2);
        const float r3n = __builtin_amdgcn_rsqf(s3);
        for (int i = 0; i < N_; ++i) {
            float4 v = *(const float4*)(X + (size_t)i * D_ + d);
            v.x *= r0n; v.y *= r1n; v.z *= r2n; v.w *= r3n;
            *(float4*)(X + (size_t)i * D_ + d) = v;
        }
    }
}

extern "C" void kernel_launch(void* const* d_in, const int* in_sizes, int n_in,
                              void* d_out, int out_size, void* d_ws, size_t ws_size,
                              hipStream_t stream) {
    (void)in_sizes; (void)n_in; (void)d_ws; (void)ws_size; (void)out_size;
    const float* x = (const float*)d_in[0];
    float* out = (float*)d_out;

    const size_t lds_bytes =
        (BLK * D_            // Xblk
         + BLK * N_          // Tt
         + NWAVES * 256      // Sred
         + N_                // norms2
         + N_                // rnorms2
         + BLK               // coeff
         + NWAVES)           // wsum
        * sizeof(float);     // = 83,040 bytes (< 320 KB/WGP)

    dim3 grid(B_ * C_);      // 256 workgroups, one per (b,c)
    dim3 block(NTHREADS);    // 8 wave32s
    GramSchmidt_30442728194303_kernel<<<grid, block, lds_bytes, stream>>>(x, out);
}